// MultiHeadAttentionBlock_11802570129434
// MI455X (gfx1250) — compile-verified
//
#include <hip/hip_runtime.h>
#include <cstdint>

// Problem constants (match reference)
#define B_   32
#define S_   512
#define D_   1024
#define H_   16
#define DH_  64
#define HD_  1024   // H_*DH_ == D_
#define EPS_ 1e-7f

typedef __attribute__((ext_vector_type(2))) float v2f;
typedef __attribute__((ext_vector_type(8))) float v8f;
typedef __attribute__((ext_vector_type(4))) unsigned int u32x4;
typedef __attribute__((ext_vector_type(4))) int i32x4;
typedef __attribute__((ext_vector_type(8))) int i32x8;

// D = A(16x4 f32) * B(4x16 f32) + C(16x16 f32)  -- CDNA5 fp32 WMMA
__device__ __forceinline__ v8f wmma_f32(v2f a, v2f b, v8f c) {
  return __builtin_amdgcn_wmma_f32_16x16x4_f32(
      /*neg_a=*/false, a, /*neg_b=*/false, b,
      /*c_mod=*/(short)0, c, /*reuse_a=*/false, /*reuse_b=*/false);
}

// ---------------------------------------------------------------------------
// Tensor Data Mover (TDM): async 2D strided tile -> LDS with LDS row padding.
// Descriptor (D#) built once per kernel; only addresses patched per issue.
// ---------------------------------------------------------------------------
#if __has_builtin(__builtin_amdgcn_tensor_load_to_lds) && __has_builtin(__builtin_amdgcn_s_wait_tensorcnt)
#define USE_TDM 1
#else
#define USE_TDM 0
#endif

#if __has_include(<hip/amd_detail/amd_gfx1250_TDM.h>)
#define TDM_ARGS6 1
#else
#define TDM_ARGS6 0
#endif

#if USE_TDM
#define TDM_WAIT() __builtin_amdgcn_s_wait_tensorcnt(0)

struct TdmDesc { u32x4 g0; i32x8 g1; i32x4 g2; i32x4 g3; };

// pad codes per D#: interval code i -> pad after 2^(i+1) dwords; amount code a -> a+1 dwords
__device__ __forceinline__ TdmDesc tdm_make_2d(uint32_t tile_w, uint32_t tile_h,
                                               uint32_t row_stride_elems,
                                               uint32_t pad_int_code,
                                               uint32_t pad_amt_code) {
  const uint32_t TD0 = 0x40000000u, TD1 = 0x40000000u;  // huge tensor dims: no OOB clamp
  TdmDesc d;
  d.g0[0] = 1u;                                  // count=1, user descriptor
  d.g0[1] = 0u; d.g0[2] = 0u; d.g0[3] = 0u;      // patched per issue
  uint32_t w0 = (2u << 16)                       // data_size = 4 bytes
              | (1u << 20)                       // pad_enable
              | (pad_int_code << 22) | (pad_amt_code << 25);
  d.g1[0] = (int)w0;
  d.g1[1] = (int)((TD0 & 0xFFFFu) << 16);        // abarrier=0 | tensor_dim0 lo16
  d.g1[2] = (int)(((TD0 >> 16) & 0xFFFFu) | ((TD1 & 0xFFFFu) << 16));
  d.g1[3] = (int)(((TD1 >> 16) & 0xFFFFu) | (tile_w << 16));   // dim1 hi | tile_dim0
  d.g1[4] = (int)(tile_h & 0xFFFFu);             // tile_dim1 | tile_dim2=0
  d.g1[5] = (int)row_stride_elems;               // tensor_dim0_stride lo32
  d.g1[6] = 0;
  d.g1[7] = 0;
  d.g2[0] = d.g2[1] = d.g2[2] = d.g2[3] = 0;
  d.g3[0] = d.g3[1] = d.g3[2] = d.g3[3] = 0;
  return d;
}

__device__ __forceinline__ void tdm_issue(TdmDesc d, const void* lds_p, const void* gptr) {
  uint64_t ga = (uint64_t)(uintptr_t)gptr;
  d.g0[1] = (uint32_t)(uintptr_t)lds_p;          // low 32 bits of flat LDS ptr = LDS offset
  d.g0[2] = (uint32_t)ga;
  d.g0[3] = (uint32_t)((ga >> 32) & 0x01FFFFFFu) | (2u << 30);  // addr[56:32] | type=2
#if TDM_ARGS6
  i32x8 gz8; for (int i = 0; i < 8; ++i) gz8[i] = 0;
  __builtin_amdgcn_tensor_load_to_lds(d.g0, d.g1, d.g2, d.g3, gz8, 0);
#else
  __builtin_amdgcn_tensor_load_to_lds(d.g0, d.g1, d.g2, d.g3, 0);
#endif
}
#endif  // USE_TDM

// ---------------------------------------------------------------------------
// Concrete dropout (float4 vectorized)
// ---------------------------------------------------------------------------
__global__ __launch_bounds__(256)
void concrete_dropout_kernel(const float4* __restrict__ x,
                             const float4* __restrict__ u,
                             const float* __restrict__ p_logits, int pidx,
                             float4* __restrict__ y, int n4) {
  int i = blockIdx.x * 256 + threadIdx.x;
  if (i >= n4) return;
  float pl = p_logits[pidx];
  float p  = 1.f / (1.f + expf(-pl));
  float la = logf(p + EPS_) - logf(1.f - p + EPS_);
  float scale = 1.f / (1.f - p);
  float4 uu = u[i];
  float4 xx = x[i];
  float4 rr;
  const float* up = (const float*)&uu;
  const float* xp = (const float*)&xx;
  float* rp = (float*)&rr;
#pragma unroll
  for (int c = 0; c < 4; ++c) {
    float t  = (la + logf(up[c] + EPS_) - logf(1.f - up[c] + EPS_)) * 10.f;  // / TEMP
    float dp = 1.f / (1.f + expf(-t));
    rp[c] = xp[c] * (1.f - dp) * scale;
  }
  y[i] = rr;
}

// ---------------------------------------------------------------------------
// GEMM: C[M x 1024] = A[M x 1024] @ W[1024 x 1024], optional ReLU.
// Block tile 128(M) x 128(N), K-tile 32, double-buffered LDS, TDM prefetch.
// 8 waves, each owning 2 m-tiles x 4 n-tiles (64 accumulator VGPRs): every
// B fragment feeds two WMMAs -> ~10.7 FLOP per LDS byte.
// ---------------------------------------------------------------------------
#define GK 1024
#define GN 1024
__global__ __launch_bounds__(256)
void gemm_wmma_kernel(const float* __restrict__ A, const float* __restrict__ W,
                      float* __restrict__ C, int relu) {
  __shared__ __align__(16) float As[2][128][34];   // 32 data + 2 pad dwords / row
  __shared__ __align__(16) float Bs[2][32][136];   // 128 data + 8 pad dwords / row
  const int tid  = threadIdx.x;
  const int wave = tid >> 5, lane = tid & 31;
  const int mg = (wave & 3) * 2;   // first of two m-tiles
  const int nw = wave >> 2;        // n-half 0..1
  const int lm = lane & 15;
  const int lk = (lane >> 4) << 1;
  const int rowBase = blockIdx.x * 128;
  const int colBase = blockIdx.y * 128;

  v8f acc[2][4];
#pragma unroll
  for (int mi = 0; mi < 2; ++mi)
#pragma unroll
    for (int i = 0; i < 4; ++i) acc[mi][i] = (v8f){0,0,0,0,0,0,0,0};

#if USE_TDM
  TdmDesc dA = tdm_make_2d(/*w=*/32,  /*h=*/128, /*stride=*/GK, /*int=*/4, /*amt=*/1);
  TdmDesc dB = tdm_make_2d(/*w=*/128, /*h=*/32,  /*stride=*/GN, /*int=*/6, /*amt=*/7);
  if (tid < 32) {
    tdm_issue(dA, &As[0][0][0], A + (size_t)rowBase * GK);
    tdm_issue(dB, &Bs[0][0][0], W + colBase);
    TDM_WAIT();
  }
  __syncthreads();
#endif

  for (int k0 = 0; k0 < GK; k0 += 32) {
    const int cur = (k0 >> 5) & 1;
#if USE_TDM
    if (tid < 32 && (k0 + 32) < GK) {   // prefetch next K-tile into alt buffer
      tdm_issue(dA, &As[cur ^ 1][0][0], A + (size_t)rowBase * GK + k0 + 32);
      tdm_issue(dB, &Bs[cur ^ 1][0][0], W + (size_t)(k0 + 32) * GN + colBase);
    }
#else
    __syncthreads();
#pragma unroll
    for (int j = 0; j < 16; ++j) {      // 128x32 A tile
      int idx = j * 256 + tid;
      As[cur][idx >> 5][idx & 31] =
          A[(size_t)(rowBase + (idx >> 5)) * GK + k0 + (idx & 31)];
    }
#pragma unroll
    for (int j = 0; j < 16; ++j) {      // 32x128 B tile
      int idx = j * 256 + tid;
      Bs[cur][idx >> 7][idx & 127] =
          W[(size_t)(k0 + (idx >> 7)) * GN + colBase + (idx & 127)];
    }
    __syncthreads();
#endif
#pragma unroll
    for (int kk = 0; kk < 32; kk += 4) {
      int kb = kk + lk;
      v2f af0 = *(const v2f*)&As[cur][(mg + 0) * 16 + lm][kb];   // aligned b64
      v2f af1 = *(const v2f*)&As[cur][(mg + 1) * 16 + lm][kb];
#pragma unroll
      for (int nt = 0; nt < 4; ++nt) {
        int n = nw * 64 + nt * 16 + lm;
        v2f bf;
        bf.x = Bs[cur][kb][n];
        bf.y = Bs[cur][kb + 1][n];
        acc[0][nt] = wmma_f32(af0, bf, acc[0][nt]);
        acc[1][nt] = wmma_f32(af1, bf, acc[1][nt]);
      }
    }
#if USE_TDM
    if (tid < 32) TDM_WAIT();   // drain prefetch before publishing alt buffer
    __syncthreads();
#endif
  }
#if !USE_TDM
  __syncthreads();
#endif
#pragma unroll
  for (int mi = 0; mi < 2; ++mi) {
#pragma unroll
    for (int nt = 0; nt < 4; ++nt) {
      int col = colBase + nw * 64 + nt * 16 + lm;
#pragma unroll
      for (int r = 0; r < 8; ++r) {
        int row = rowBase + (mg + mi) * 16 + r + ((lane >> 4) << 3);
        float v = acc[mi][nt][r];
        float* dst = &C[(size_t)row * GN + col];
        if (relu) {
          v = v > 0.f ? v : 0.f;
          __builtin_nontemporal_store(v, dst);   // final output: never re-read
        } else {
          *dst = v;                              // Q/K/V: consumed from L2
        }
      }
    }
  }
}

// ---------------------------------------------------------------------------
// Attention: per block (b, h, 32 query rows).
//   E = Q K^T / 8 (masked) -> softmax in LDS -> alpha to d_out -> out = alpha @ V
// K/V streamed in 128-key chunks, TDM double-buffered; first V chunk DMA is
// hidden behind the softmax pass.
// ---------------------------------------------------------------------------
__global__ __launch_bounds__(256)
void attention_kernel(const float* __restrict__ Q, const float* __restrict__ K,
                      const float* __restrict__ V, const int* __restrict__ lengths,
                      float* __restrict__ alpha_out, float* __restrict__ attn_out) {
  __shared__ __align__(16) float Es[32][514];       // scores / alpha
  __shared__ __align__(16) float Qs[32][66];
  __shared__ __align__(16) float KVs[2][128][66];   // K (phase 1) / V (phase 2)
  __shared__ float red[32][8];

  const int q0  = blockIdx.x * 32;
  const int h   = blockIdx.y;
  const int b   = blockIdx.z;
  const int len = lengths[b];
  const int tid = threadIdx.x, wave = tid >> 5, lane = tid & 31;
  const int lm = lane & 15, lg = lane >> 4, lk = lg << 1;
  const size_t base = (size_t)b * S_ * HD_ + (size_t)h * DH_;  // + s*HD_ + d

#if USE_TDM
  TdmDesc dQ  = tdm_make_2d(/*w=*/64, /*h=*/32,  /*stride=*/HD_, /*int=*/5, /*amt=*/1);
  TdmDesc dKV = tdm_make_2d(/*w=*/64, /*h=*/128, /*stride=*/HD_, /*int=*/5, /*amt=*/1);
  if (tid < 32) {
    tdm_issue(dQ,  &Qs[0][0],     Q + base + (size_t)q0 * HD_);
    tdm_issue(dKV, &KVs[0][0][0], K + base);     // key chunk 0
    TDM_WAIT();
  }
  __syncthreads();
#else
#pragma unroll
  for (int j = 0; j < 8; ++j) {
    int idx = j * 256 + tid;
    Qs[idx >> 6][idx & 63] = Q[base + (size_t)(q0 + (idx >> 6)) * HD_ + (idx & 63)];
  }
#endif

  // ----- Phase 1: E = Q K^T * 0.125, masked -----
  const int mt = wave & 1;
  const int nb = wave >> 1;
  for (int kc = 0; kc < 4; ++kc) {
    const int cur = kc & 1;
#if USE_TDM
    if (tid < 32 && kc < 3)   // prefetch next key chunk
      tdm_issue(dKV, &KVs[cur ^ 1][0][0], K + base + (size_t)((kc + 1) * 128) * HD_);
#else
    __syncthreads();
#pragma unroll
    for (int j = 0; j < 32; ++j) {
      int idx = j * 256 + tid;
      KVs[cur][idx >> 6][idx & 63] =
          K[base + (size_t)(kc * 128 + (idx >> 6)) * HD_ + (idx & 63)];
    }
    __syncthreads();
#endif
    v8f acc[2];
    acc[0] = (v8f){0,0,0,0,0,0,0,0};
    acc[1] = (v8f){0,0,0,0,0,0,0,0};
#pragma unroll
    for (int kk = 0; kk < 64; kk += 4) {   // head dim
      int kb = kk + lk;
      v2f af = *(const v2f*)&Qs[mt * 16 + lm][kb];
#pragma unroll
      for (int nt = 0; nt < 2; ++nt) {
        int keyl = nb * 32 + nt * 16 + lm;
        v2f bf = *(const v2f*)&KVs[cur][keyl][kb];   // B[k=d][n=key] = K_chunk[key][d]
        acc[nt] = wmma_f32(af, bf, acc[nt]);
      }
    }
#pragma unroll
    for (int nt = 0; nt < 2; ++nt) {
      int keyl = nb * 32 + nt * 16 + lm;
      int key  = kc * 128 + keyl;
      bool kok = key <= len;
#pragma unroll
      for (int r = 0; r < 8; ++r) {
        int m = mt * 16 + r + (lg << 3);
        bool valid = kok && ((q0 + m) <= len);
        Es[m][key] = valid ? acc[nt][r] * 0.125f : -1e10f;
      }
    }
#if USE_TDM
    if (tid < 32) TDM_WAIT();
    __syncthreads();
#endif
  }
  __syncthreads();

#if USE_TDM
  if (tid < 32)   // hide first V-chunk DMA behind softmax
    tdm_issue(dKV, &KVs[0][0][0], V + base);
#endif

  // ----- Softmax per row (8 threads x 64 cols per row) -----
  {
    int row = tid >> 3, seg = tid & 7;
    int c0 = seg * 64;
    float mx = -INFINITY;
    for (int c = c0; c < c0 + 64; ++c) mx = fmaxf(mx, Es[row][c]);
    red[row][seg] = mx;
    __syncthreads();
    mx = red[row][0];
#pragma unroll
    for (int i = 1; i < 8; ++i) mx = fmaxf(mx, red[row][i]);
    __syncthreads();
    float sum = 0.f;
    for (int c = c0; c < c0 + 64; ++c) {
      float e = expf(Es[row][c] - mx);
      Es[row][c] = e;
      sum += e;
    }
    red[row][seg] = sum;
    __syncthreads();
    sum = 0.f;
#pragma unroll
    for (int i = 0; i < 8; ++i) sum += red[row][i];
    float inv = 1.f / sum;
    size_t abase = (((size_t)b * H_ + h) * S_ + (q0 + row)) * S_;
    for (int c = c0; c < c0 + 64; ++c) {
      float a = Es[row][c] * inv;
      Es[row][c] = a;
      __builtin_nontemporal_store(a, &alpha_out[abase + c]);  // streamed, never re-read
    }
  }
#if USE_TDM
  if (tid < 32) TDM_WAIT();
#endif
  __syncthreads();   // alpha visible to all waves (+ V chunk 0 ready)

  // ----- Phase 2: out = alpha @ V (M=32, K=512, N=64) -----
  const int mt2 = wave & 1;
  const int nt2 = wave >> 1;
  v8f oacc = (v8f){0,0,0,0,0,0,0,0};
  for (int kc = 0; kc < 4; ++kc) {
    const int cur = kc & 1;
#if USE_TDM
    if (tid < 32 && kc < 3)   // prefetch next V chunk
      tdm_issue(dKV, &KVs[cur ^ 1][0][0], V + base + (size_t)((kc + 1) * 128) * HD_);
#else
    if (kc) __syncthreads();
#pragma unroll
    for (int j = 0; j < 32; ++j) {
      int idx = j * 256 + tid;
      KVs[cur][idx >> 6][idx & 63] =
          V[base + (size_t)(kc * 128 + (idx >> 6)) * HD_ + (idx & 63)];
    }
    __syncthreads();
#endif
#pragma unroll
    for (int kk = 0; kk < 128; kk += 4) {   // keys in chunk
      int kb = kk + lk;
      v2f af = *(const v2f*)&Es[mt2 * 16 + lm][kc * 128 + kb];
      int n = nt2 * 16 + lm;
      v2f bf;                     // B[k=key][n=d] = V_chunk[key][d]
      bf.x = KVs[cur][kb][n];
      bf.y = KVs[cur][kb + 1][n];
      oacc = wmma_f32(af, bf, oacc);
    }
#if USE_TDM
    if (tid < 32) TDM_WAIT();
    __syncthreads();
#endif
  }
  {
    int d = nt2 * 16 + lm;
#pragma unroll
    for (int r = 0; r < 8; ++r) {
      int m = mt2 * 16 + r + (lg << 3);
      attn_out[base + (size_t)(q0 + m) * HD_ + d] = oacc[r];
    }
  }
}

// ---------------------------------------------------------------------------
// Launch
// ---------------------------------------------------------------------------
extern "C" void kernel_launch(void* const* d_in, const int* in_sizes, int n_in,
                              void* d_out, int out_size, void* d_ws, size_t ws_size,
                              hipStream_t stream) {
  const float* x       = (const float*)d_in[0];
  const int*   lengths = (const int*)d_in[1];
  const float* Wq      = (const float*)d_in[2];
  const float* Wk      = (const float*)d_in[3];
  const float* Wv      = (const float*)d_in[4];
  const float* Wo      = (const float*)d_in[5];
  const float* plog    = (const float*)d_in[6];
  const float* nq      = (const float*)d_in[7];
  const float* nk      = (const float*)d_in[8];
  const float* nv      = (const float*)d_in[9];
  const float* no      = (const float*)d_in[10];

  float* out   = (float*)d_out;                    // [B,S,D] relu output
  float* alpha = out + (size_t)B_ * S_ * D_;       // [B,H,S,S]

  const size_t NE = (size_t)B_ * S_ * D_;          // 16,777,216 elements
  float* ws = (float*)d_ws;
  float* dx = ws;            // dropout scratch, later attention output
  float* Qb = ws + NE;
  float* Kb = ws + 2 * NE;
  float* Vb = ws + 3 * NE;

  const int M = B_ * S_;                           // 16384
  dim3 gGemm(M / 128, GN / 128);                   // (128, 8)
  dim3 gDrop((unsigned)(NE / 4 / 256));            // float4 kernel
  dim3 gAttn(S_ / 32, H_, B_);                     // (16, 16, 32)
  const int N4 = (int)(NE / 4);

  concrete_dropout_kernel<<<gDrop, 256, 0, stream>>>(
      (const float4*)x, (const float4*)nq, plog, 0, (float4*)dx, N4);
  gemm_wmma_kernel<<<gGemm, 256, 0, stream>>>(dx, Wq, Qb, 0);
  concrete_dropout_kernel<<<gDrop, 256, 0, stream>>>(
      (const float4*)x, (const float4*)nk, plog, 1, (float4*)dx, N4);
  gemm_wmma_kernel<<<gGemm, 256, 0, stream>>>(dx, Wk, Kb, 0);
  concrete_dropout_kernel<<<gDrop, 256, 0, stream>>>(
      (const float4*)x, (const float4*)nv, plog, 2, (float4*)dx, N4);
  gemm_wmma_kernel<<<gGemm, 256, 0, stream>>>(dx, Wv, Vb, 0);
  attention_kernel<<<gAttn, 256, 0, stream>>>(Qb, Kb, Vb, lengths, alpha, dx);
  concrete_dropout_kernel<<<gDrop, 256, 0, stream>>>(
      (const float4*)dx, (const float4*)no, plog, 3, (float4*)Qb, N4);
  gemm_wmma_kernel<<<gGemm, 256, 0, stream>>>(Qb, Wo, out, 1);
}